// STACK_FF_NLP_15324443312625
// MI455X (gfx1250) — compile-verified
//
#include <hip/hip_runtime.h>

#define I_SIZE 4096
#define H_SIZE 2048
#define WIN 10
#define NB 6
#define EPSV 1e-9f
#define CATW (2*I_SIZE)   // 8192, row stride of concatenated [invec|hvec]

// ---------------- workspace layout (float offsets) ----------------
#define OF_HCOEF   0                                  // [NB][H] stack-read coefs for hvec
#define OF_ICOEF0  (OF_HCOEF + NB*H_SIZE)             // [H]     stack-read coefs for invec (branch 0)
#define OF_DCOEF   (OF_ICOEF0 + H_SIZE)               // [NB][H] diagonal scale of stackmem
#define OF_QCOEF   (OF_DCOEF + NB*H_SIZE)             // [NB][H] rank-1 coef for new_inp push
#define OF_RCOEF   (OF_QCOEF + NB*H_SIZE)             // [NB][WIN][H] rank-1 coefs mapped to input rows
#define OF_SPTR    (OF_RCOEF + NB*WIN*H_SIZE)         // [NB][H] sptrmem
#define OF_IPTR    (OF_SPTR + NB*H_SIZE)              // [NB][11] iptrmem (reserve 96)
#define OF_HIN     (OF_IPTR + 96)                     // [NB][WIN] input-row mix into hvec
#define OF_IIN     (OF_HIN + 64)                      // [NB][WIN] input-row mix into invec
#define OF_IAVAIL  (OF_IIN + 64)                      // [NB] input_avail
#define OF_SCAL    (OF_IAVAIL + 8)                    // pp[6] Sh[6] Si[6] kldh[6] kldi[6] w[6]
#define OF_XC      (OF_SCAL + 64)                     // [16][8192]  rows 0..5 = [Xi_b | Xh_b], 6..15 zero
#define OF_L1      (OF_XC + 16*CATW)                  // [NB][I] logits (ff, later kouth)
#define OF_L2      (OF_L1 + NB*I_SIZE)                // [NB][I] logits (kin, later kouti)
#define OF_NEWI    (OF_L2 + NB*I_SIZE)                // [16][I] new_inp (rows 6..15 zero)
#define OF_SEFF    (OF_NEWI + 16*I_SIZE)              // [H]
#define OF_REFF    (OF_SEFF + H_SIZE)                 // [WIN][H]
#define OF_QEFF    (OF_REFF + WIN*H_SIZE)             // [NB][H]

typedef __attribute__((ext_vector_type(2))) float v2f;
typedef __attribute__((ext_vector_type(8))) float v8f;

// ---------------- block reductions ----------------
template<int BS>
__device__ __forceinline__ float bsum(float v, float* red) {
  int t = threadIdx.x;
  red[t] = v; __syncthreads();
  for (int s = BS/2; s > 0; s >>= 1) {
    if (t < s) red[t] += red[t+s];
    __syncthreads();
  }
  float r = red[0]; __syncthreads();
  return r;
}
template<int BS>
__device__ __forceinline__ float bmaxr(float v, float* red) {
  int t = threadIdx.x;
  red[t] = v; __syncthreads();
  for (int s = BS/2; s > 0; s >>= 1) {
    if (t < s) red[t] = fmaxf(red[t], red[t+s]);
    __syncthreads();
  }
  float r = red[0]; __syncthreads();
  return r;
}

// ---------------- pointer algebra: reduce all push/pop/get to [H] coefficient vectors ----------------
__global__ __launch_bounds__(256) void k_pointer(const float* __restrict__ sp0,
                                                 const float* __restrict__ ip0,
                                                 float* __restrict__ ws)
{
  __shared__ float sp[H_SIZE], spn[H_SIZE], D[H_SIZE];
  __shared__ float C[4][H_SIZE];
  __shared__ float red[256];
  __shared__ float ipSh[12], ipTmp[12];
  __shared__ float gSh[4][WIN];
  int t = threadIdx.x;
  float* hcoef  = ws + OF_HCOEF;
  float* icoef0 = ws + OF_ICOEF0;
  float* dcoef  = ws + OF_DCOEF;
  float* qcoef  = ws + OF_QCOEF;
  float* rcoef  = ws + OF_RCOEF;
  float* sptr   = ws + OF_SPTR;
  float* iptr   = ws + OF_IPTR;
  float* hin    = ws + OF_HIN;
  float* iin    = ws + OF_IIN;
  float* iavail = ws + OF_IAVAIL;

  // ---- branch 0: pop, pop, calforward, push ----
  for (int h=t; h<H_SIZE; h+=256) sp[h] = sp0[h];
  __syncthreads();
  for (int h=t; h<H_SIZE; h+=256) spn[h] = (h < H_SIZE-1) ? sp[h+1] : 0.f;   // pop1 shift-up
  __syncthreads();
  float part = 0.f;
  for (int h=t; h<H_SIZE; h+=256) part += spn[h];
  float s = bsum<256>(part, red);
  float f = (s > 0.f) ? 1.f/s : 1.f;
  for (int h=t; h<H_SIZE; h+=256) { icoef0[h] = spn[h]*f; D[h] = 1.f - spn[h]; }
  __syncthreads();
  if (t == 0) spn[0] += sp[0];
  __syncthreads();
  for (int h=t; h<H_SIZE; h+=256) sp[h] = (h < H_SIZE-1) ? spn[h+1] : 0.f;   // pop2 shift-up
  __syncthreads();
  part = 0.f;
  for (int h=t; h<H_SIZE; h+=256) part += sp[h];
  s = bsum<256>(part, red);
  f = (s > 0.f) ? 1.f/s : 1.f;
  for (int h=t; h<H_SIZE; h+=256) { hcoef[h] = sp[h]*f*D[h]; D[h] *= (1.f - sp[h]); }
  __syncthreads();
  if (t == 0) sp[0] += spn[0];
  __syncthreads();
  for (int h=t; h<H_SIZE; h+=256) {                                          // push new_inp
    float p = sp[h];
    qcoef[h] = p;
    dcoef[h] = D[h]*(1.f - p);
    sptr[h]  = (h > 0) ? sp[h-1] : 0.f;
  }
  for (int m=t; m<WIN; m+=256) { hin[m] = 0.f; iin[m] = 0.f; }
  for (int idx=t; idx<WIN*H_SIZE; idx+=256) rcoef[idx] = 0.f;
  for (int m=t; m<11; m+=256) iptr[m] = ip0[m];
  if (t == 0) iavail[0] = 1.f;
  __syncthreads();

  // ---- branches 1..5 ----
  for (int b=1; b<NB; ++b) {
    for (int h=t; h<H_SIZE; h+=256) { sp[h] = sp0[h]; D[h] = 1.f; }
    if (t < 11) ipSh[t] = ip0[t];
    __syncthreads();
    int np = b - 1;
    for (int k=0; k<np; ++k) {            // get + push, (b-1) times
      if (t < WIN) gSh[k][t] = ipSh[t];
      __syncthreads();
      if (t < 11) ipTmp[t] = (t == 0) ? 0.f : (ipSh[t-1] + ((t == 10) ? ipSh[10] : 0.f));
      __syncthreads();
      if (t < 11) ipSh[t] = ipTmp[t];
      for (int h=t; h<H_SIZE; h+=256) {
        float p = sp[h], om = 1.f - p;
        D[h] *= om;
        for (int j=0; j<k; ++j) C[j][h] *= om;
        C[k][h] = p;
      }
      __syncthreads();
      for (int h=t; h<H_SIZE; h+=256) spn[h] = (h > 0) ? sp[h-1] : 0.f;      // shift-down
      __syncthreads();
      for (int h=t; h<H_SIZE; h+=256) sp[h] = spn[h];
      __syncthreads();
    }
    // final get -> invec mix over input rows
    if (t < WIN) iin[b*WIN + t] = ipSh[t];
    __syncthreads();
    if (t < 11) ipTmp[t] = (t == 0) ? 0.f : (ipSh[t-1] + ((t == 10) ? ipSh[10] : 0.f));
    __syncthreads();
    if (t < 11) ipSh[t] = ipTmp[t];
    __syncthreads();
    // pop -> hvec coefficients
    for (int h=t; h<H_SIZE; h+=256) spn[h] = (h < H_SIZE-1) ? sp[h+1] : 0.f;
    __syncthreads();
    part = 0.f;
    for (int h=t; h<H_SIZE; h+=256) part += spn[h];
    s = bsum<256>(part, red);
    f = (s > 0.f) ? 1.f/s : 1.f;
    for (int h=t; h<H_SIZE; h+=256) hcoef[b*H_SIZE+h] = spn[h]*f*D[h];
    float dj[4] = {0.f,0.f,0.f,0.f};
    for (int j=0; j<np; ++j) {
      part = 0.f;
      for (int h=t; h<H_SIZE; h+=256) part += spn[h]*f*C[j][h];
      dj[j] = bsum<256>(part, red);
    }
    if (t < WIN) {
      float a = 0.f;
      for (int j=0; j<np; ++j) a += dj[j]*gSh[j][t];
      hin[b*WIN + t] = a;
    }
    for (int h=t; h<H_SIZE; h+=256) {
      float om = 1.f - spn[h];
      D[h] *= om;
      for (int j=0; j<np; ++j) C[j][h] *= om;
    }
    __syncthreads();
    if (t == 0) spn[0] += sp[0];
    __syncthreads();
    for (int h=t; h<H_SIZE; h+=256) sp[h] = spn[h];
    __syncthreads();
    // push new_inp
    for (int h=t; h<H_SIZE; h+=256) {
      float p = sp[h], om = 1.f - p;
      qcoef[b*H_SIZE+h] = p;
      dcoef[b*H_SIZE+h] = D[h]*om;
      for (int j=0; j<np; ++j) C[j][h] *= om;
    }
    __syncthreads();
    for (int h=t; h<H_SIZE; h+=256) sptr[b*H_SIZE+h] = (h > 0) ? sp[h-1] : 0.f;
    for (int h=t; h<H_SIZE; h+=256) {
      for (int m=0; m<WIN; ++m) {
        float a = 0.f;
        for (int j=0; j<np; ++j) a += C[j][h]*gSh[j][m];
        rcoef[(b*WIN + m)*H_SIZE + h] = a;
      }
    }
    if (t < 11) iptr[b*11 + t] = ipSh[t];
    if (t == 0) iavail[b] = 1.f - ipSh[10];
    __syncthreads();
  }
}

// ---------------- invec/hvec extraction: one streaming pass over the stack ----------------
// Writes concatenated XC[16][8192] = [invec | hvec] per branch row, pad rows zeroed.
// Also zeroes NEWI pad rows 6..15 so the later GEMMs are branch-free.
__global__ __launch_bounds__(256) void k_vec(const float* __restrict__ stackM,
                                             const float* __restrict__ input,
                                             float* __restrict__ ws)
{
  const float* hcoef  = ws + OF_HCOEF;
  const float* icoef0 = ws + OF_ICOEF0;
  const float* hin    = ws + OF_HIN;
  const float* iin    = ws + OF_IIN;
  float* Xc   = ws + OF_XC;
  float* newi = ws + OF_NEWI;
  int t = threadIdx.x;
  int i = blockIdx.x * 256 + t;
  __shared__ float sc[7*128];
  float ah[NB] = {0.f,0.f,0.f,0.f,0.f,0.f};
  float ai[NB] = {0.f,0.f,0.f,0.f,0.f,0.f};
  for (int h0 = 0; h0 < H_SIZE; h0 += 128) {
    for (int idx = t; idx < 7*128; idx += 256) {
      int row = idx >> 7, hh = idx & 127;
      sc[idx] = (row < NB) ? hcoef[row*H_SIZE + h0 + hh] : icoef0[h0 + hh];
    }
    __syncthreads();
    for (int hh = 0; hh < 128; ++hh) {
      float sv = stackM[(size_t)(h0 + hh) * I_SIZE + i];
      #pragma unroll
      for (int b = 0; b < NB; ++b) ah[b] += sc[b*128 + hh] * sv;
      ai[0] += sc[6*128 + hh] * sv;
    }
    __syncthreads();
  }
  for (int m = 0; m < WIN; ++m) {
    float iv = input[m * I_SIZE + i];
    #pragma unroll
    for (int b = 0; b < NB; ++b) {
      ah[b] += hin[b*WIN + m] * iv;
      ai[b] += iin[b*WIN + m] * iv;
    }
  }
  #pragma unroll
  for (int b = 0; b < NB; ++b) {
    Xc[(size_t)b*CATW + i]          = ai[b];
    Xc[(size_t)b*CATW + I_SIZE + i] = ah[b];
  }
  #pragma unroll
  for (int b = NB; b < 16; ++b) {                 // zero pad rows for WMMA
    Xc[(size_t)b*CATW + i] = 0.f;
    Xc[(size_t)b*CATW + I_SIZE + i] = 0.f;
    newi[(size_t)b*I_SIZE + i] = 0.f;
  }
}

// ---------------- batched skinny GEMM via V_WMMA_F32_16X16X4_F32 ----------------
// out[m][n] = sum_k A[m][k] * W[n][k] + bias[n], m<6 (A zero-padded to 16 rows).
// One block = one 16-column tile; 8 waves split K; LDS reduction (deterministic order).
__global__ __launch_bounds__(256) void k_gemm16(const float* __restrict__ A, int lda,
                                                int K,
                                                const float* __restrict__ W,
                                                const float* __restrict__ bias,
                                                float* __restrict__ out)
{
  __shared__ float lred[8*256];
  int lane = threadIdx.x & 31;
  int wid  = threadIdx.x >> 5;          // K-chunk id, 0..7
  int n0   = blockIdx.x * 16;
  int m    = lane & 15;                 // A row / B column
  int half = lane >> 4;                 // K pair {0,1} vs {2,3}
  int KC   = K >> 3;                    // K per chunk (512 or 1024)
  int kbase = wid * KC;
  const float* arow = A + (size_t)m * lda        + kbase + 2*half;
  const float* wrow = W + (size_t)(n0 + m) * K   + kbase + 2*half;

  v8f acc = {0.f,0.f,0.f,0.f,0.f,0.f,0.f,0.f};
  v2f a[8], b[8];
  #pragma unroll
  for (int u = 0; u < 8; ++u) {
    a[u] = *(const v2f*)(arow + u*4);
    b[u] = *(const v2f*)(wrow + u*4);
  }
  for (int kg = 32; kg < KC; kg += 32) {
    v2f a2[8], b2[8];
    #pragma unroll
    for (int u = 0; u < 8; ++u) {                  // prefetch next group
      a2[u] = *(const v2f*)(arow + kg + u*4);
      b2[u] = *(const v2f*)(wrow + kg + u*4);
    }
    #pragma unroll
    for (int u = 0; u < 8; ++u)                    // consume current group
      acc = __builtin_amdgcn_wmma_f32_16x16x4_f32(false, a[u], false, b[u],
                                                  (short)0, acc, false, false);
    #pragma unroll
    for (int u = 0; u < 8; ++u) { a[u] = a2[u]; b[u] = b2[u]; }
  }
  #pragma unroll
  for (int u = 0; u < 8; ++u)
    acc = __builtin_amdgcn_wmma_f32_16x16x4_f32(false, a[u], false, b[u],
                                                (short)0, acc, false, false);

  // cross-wave (split-K) reduction through LDS, fixed order -> deterministic
  #pragma unroll
  for (int r = 0; r < 8; ++r) lred[wid*256 + lane*8 + r] = acc[r];
  __syncthreads();
  int t = threadIdx.x;
  int lane_ = t >> 3, r_ = t & 7;
  float v = 0.f;
  #pragma unroll
  for (int wdx = 0; wdx < 8; ++wdx) v += lred[wdx*256 + t];
  int row = r_ + ((lane_ >> 4) << 3);
  int col = n0 + (lane_ & 15);
  if (row < NB) out[(size_t)row * I_SIZE + col] = v + bias[col];
}

// ---------------- pred path: log_softmax + KLD(predvec, invec) + raw sums ----------------
__global__ __launch_bounds__(1024) void k_pred_kld(float* __restrict__ ws)
{
  __shared__ float red[1024];
  int b = blockIdx.x, t = threadIdx.x;
  const float* z  = ws + OF_L1 + (size_t)b*I_SIZE;
  const float* qi = ws + OF_XC + (size_t)b*CATW;            // invec
  const float* qh = qi + I_SIZE;                            // hvec
  float zv[4], qv[4];
  float mx = -3.0e38f;
  #pragma unroll
  for (int j=0;j<4;++j){ zv[j] = z[t + j*1024]; mx = fmaxf(mx, zv[j]); }
  mx = bmaxr<1024>(mx, red);
  float se = 0.f;
  #pragma unroll
  for (int j=0;j<4;++j) se += __expf(zv[j]-mx);
  se = bsum<1024>(se, red);
  float lse = mx + __logf(se);
  float P=0.f, Q=0.f, Si=0.f, Sh=0.f;
  #pragma unroll
  for (int j=0;j<4;++j){
    zv[j] -= lse;                         // log-softmax
    qv[j] = qi[t + j*1024];
    float hvv = qh[t + j*1024];
    P += zv[j] + EPSV; Q += qv[j] + EPSV; Si += qv[j]; Sh += hvv;
  }
  P = bsum<1024>(P, red); Q = bsum<1024>(Q, red);
  Si = bsum<1024>(Si, red); Sh = bsum<1024>(Sh, red);
  float iP = 1.f/P, iQ = 1.f/Q, a = 0.f;
  #pragma unroll
  for (int j=0;j<4;++j){
    float r  = (zv[j]+EPSV)*iP;
    float qn = (qv[j]+EPSV)*iQ;
    a += r*__logf(r/qn);
  }
  a = bsum<1024>(a, red);
  if (t == 0) { float* sc = ws + OF_SCAL; sc[b] = a; sc[6+b] = Sh; sc[12+b] = Si; }
}

// ---------------- kin path: softmax -> new_inp ----------------
__global__ __launch_bounds__(1024) void k_softmax_newinp(float* __restrict__ ws)
{
  __shared__ float red[1024];
  int b = blockIdx.x, t = threadIdx.x;
  const float* z = ws + OF_L2 + (size_t)b*I_SIZE;
  float* o = ws + OF_NEWI + (size_t)b*I_SIZE;
  float zv[4]; float mx = -3.0e38f;
  #pragma unroll
  for (int j=0;j<4;++j){ zv[j] = z[t + j*1024]; mx = fmaxf(mx, zv[j]); }
  mx = bmaxr<1024>(mx, red);
  float se = 0.f;
  #pragma unroll
  for (int j=0;j<4;++j) se += __expf(zv[j]-mx);
  se = bsum<1024>(se, red);
  float lse = mx + __logf(se);
  #pragma unroll
  for (int j=0;j<4;++j) o[t + j*1024] = __expf(zv[j]-lse);
}

// ---------------- proj paths: softmax + KLD vs hvec (blocks 0-5) / invec (blocks 6-11) ----------------
__global__ __launch_bounds__(1024) void k_proj_kld(float* __restrict__ ws)
{
  __shared__ float red[1024];
  int idx = blockIdx.x; int b = idx % NB; bool isH = (idx < NB);
  int t = threadIdx.x;
  const float* z = ws + (isH ? OF_L1 : OF_L2) + (size_t)b*I_SIZE;   // L1/L2 reused by kouth/kouti
  const float* q = ws + OF_XC + (size_t)b*CATW + (isH ? I_SIZE : 0);
  float zv[4], qv[4]; float mx = -3.0e38f;
  #pragma unroll
  for (int j=0;j<4;++j){ zv[j] = z[t + j*1024]; mx = fmaxf(mx, zv[j]); }
  mx = bmaxr<1024>(mx, red);
  float se = 0.f;
  #pragma unroll
  for (int j=0;j<4;++j) se += __expf(zv[j]-mx);
  se = bsum<1024>(se, red);
  float lse = mx + __logf(se);
  float P=0.f, Q=0.f;
  #pragma unroll
  for (int j=0;j<4;++j){
    zv[j] = __expf(zv[j]-lse);            // softmax
    qv[j] = q[t + j*1024];
    P += zv[j] + EPSV; Q += qv[j] + EPSV;
  }
  P = bsum<1024>(P, red); Q = bsum<1024>(Q, red);
  float iP = 1.f/P, iQ = 1.f/Q, a = 0.f;
  #pragma unroll
  for (int j=0;j<4;++j){
    float r  = (zv[j]+EPSV)*iP;
    float qn = (qv[j]+EPSV)*iQ;
    a += r*__logf(r/qn);
  }
  a = bsum<1024>(a, red);
  if (t == 0) ws[OF_SCAL + (isH ? 18 : 24) + b] = a;
}

// ---------------- gate, wldl recurrence, w-normalize, new_sp/new_ip, effective coefficients ----------------
__global__ __launch_bounds__(256) void k_gate(const float* __restrict__ gate_w,
                                              const float* __restrict__ gate_b,
                                              float* __restrict__ ws,
                                              float* __restrict__ out)
{
  __shared__ float red[256];
  __shared__ float wsh[NB];
  int t = threadIdx.x;
  float* scal = ws + OF_SCAL;
  const float* iavail = ws + OF_IAVAIL;
  const float* sptr = ws + OF_SPTR;
  const float* iptr = ws + OF_IPTR;
  const float* dcoef = ws + OF_DCOEF;
  const float* qcoef = ws + OF_QCOEF;
  const float* rcoef = ws + OF_RCOEF;
  float* Seff = ws + OF_SEFF;
  float* Reff = ws + OF_REFF;
  float* Qeff = ws + OF_QEFF;
  if (t == 0) {
    float wl[NB], perp[NB];
    for (int b=0; b<NB; ++b) {
      float pp = scal[b];
      perp[b] = pp + 0.5f*scal[24+b] + 0.5f*scal[18+b];
      float l0 = gate_w[0]*pp + gate_b[0];
      float l1 = gate_w[1]*pp + gate_b[1];
      float mm = fmaxf(l0, l1);
      float e0 = __expf(l0 - mm), e1 = __expf(l1 - mm);
      float g1 = e1 / (e0 + e1);
      if (scal[6+b] == 0.f) g1 = 0.f;           // sum(hvec)==0 -> gate=[1,0]
      else if (scal[12+b] == 0.f) g1 = 1.f;     // sum(invec)==0 -> gate=[0,1]
      wl[b] = (b == 0) ? g1 : (1.f - wl[b-1]) * g1 * iavail[b];
    }
    float sw = 0.f;
    for (int b=0; b<NB; ++b) sw += wl[b];
    float inv = 1.f / sw;
    float resperp = 0.f;
    for (int b=0; b<NB; ++b) { wl[b] *= inv; resperp += wl[b]*perp[b]; wsh[b] = wl[b]; }
    out[0] = resperp;
  }
  __syncthreads();
  float w[NB];
  #pragma unroll
  for (int b=0; b<NB; ++b) w[b] = wsh[b];
  size_t obase = 1 + (size_t)H_SIZE * I_SIZE;
  // new_sp = normalize(sum_b w_b * sptr_b)
  float part = 0.f;
  for (int h=t; h<H_SIZE; h+=256) {
    float a = 0.f;
    #pragma unroll
    for (int b=0; b<NB; ++b) a += w[b]*sptr[b*H_SIZE+h];
    part += a;
  }
  float ssum = bsum<256>(part, red);
  float inv2 = (ssum > 0.f) ? 1.f/ssum : 1.f;
  for (int h=t; h<H_SIZE; h+=256) {
    float a = 0.f;
    #pragma unroll
    for (int b=0; b<NB; ++b) a += w[b]*sptr[b*H_SIZE+h];
    out[obase + h] = a*inv2;
  }
  // new_ip
  if (t == 0) {
    float tmp[11]; float si = 0.f;
    for (int m=0; m<11; ++m) {
      float a = 0.f;
      for (int b=0; b<NB; ++b) a += w[b]*iptr[b*11+m];
      tmp[m] = a; si += a;
    }
    float inv3 = (si > 0.f) ? 1.f/si : 1.f;
    for (int m=0; m<11; ++m) out[obase + H_SIZE + m] = tmp[m]*inv3;
  }
  // effective combine coefficients
  for (int h=t; h<H_SIZE; h+=256) {
    float se = 0.f;
    #pragma unroll
    for (int b=0; b<NB; ++b) {
      se += w[b]*dcoef[b*H_SIZE+h];
      Qeff[b*H_SIZE+h] = w[b]*qcoef[b*H_SIZE+h];
    }
    Seff[h] = se;
    for (int m=0; m<WIN; ++m) {
      float a = 0.f;
      #pragma unroll
      for (int b=0; b<NB; ++b) a += w[b]*rcoef[(b*WIN+m)*H_SIZE+h];
      Reff[m*H_SIZE+h] = a;
    }
  }
}

// ---------------- new_stack = Seff ⊙ stack + Reff^T input + Qeff^T new_inp ----------------
__global__ __launch_bounds__(256) void k_combine(const float* __restrict__ stackM,
                                                 const float* __restrict__ input,
                                                 const float* __restrict__ ws,
                                                 float* __restrict__ out)
{
  __shared__ float cs[17];
  int h = blockIdx.x, t = threadIdx.x;
  const float* Seff = ws + OF_SEFF;
  const float* Reff = ws + OF_REFF;
  const float* Qeff = ws + OF_QEFF;
  const float* newi = ws + OF_NEWI;
  if (t < 17) {
    cs[t] = (t == 0) ? Seff[h]
          : ((t < 11) ? Reff[(t-1)*H_SIZE + h] : Qeff[(t-11)*H_SIZE + h]);
  }
  __syncthreads();
  for (int i=t; i<I_SIZE; i+=256) {
    float a = cs[0]*stackM[(size_t)h*I_SIZE + i];
    #pragma unroll
    for (int m=0; m<WIN; ++m) a += cs[1+m]*input[m*I_SIZE + i];
    #pragma unroll
    for (int b=0; b<NB; ++b) a += cs[11+b]*newi[b*I_SIZE + i];
    out[1 + (size_t)h*I_SIZE + i] = a;
  }
}

extern "C" void kernel_launch(void* const* d_in, const int* in_sizes, int n_in,
                              void* d_out, int out_size, void* d_ws, size_t ws_size,
                              hipStream_t stream)
{
  (void)in_sizes; (void)n_in; (void)out_size; (void)ws_size;
  const float* input  = (const float*)d_in[0];
  const float* stackM = (const float*)d_in[1];
  const float* sp0    = (const float*)d_in[2];
  const float* ip0    = (const float*)d_in[3];
  const float* ff_w   = (const float*)d_in[4];
  const float* ff_b   = (const float*)d_in[5];
  const float* gate_w = (const float*)d_in[6];
  const float* gate_b = (const float*)d_in[7];
  const float* kin_w  = (const float*)d_in[8];
  const float* kin_b  = (const float*)d_in[9];
  const float* kouth_w= (const float*)d_in[10];
  const float* kouth_b= (const float*)d_in[11];
  const float* kouti_w= (const float*)d_in[12];
  const float* kouti_b= (const float*)d_in[13];
  float* out = (float*)d_out;
  float* ws  = (float*)d_ws;

  k_pointer<<<dim3(1), dim3(256), 0, stream>>>(sp0, ip0, ws);
  k_vec<<<dim3(I_SIZE/256), dim3(256), 0, stream>>>(stackM, input, ws);
  // ff: pred logits = hvec @ ff_w^T + ff_b -> L1   (A = hvec half of XC)
  k_gemm16<<<dim3(256), dim3(256), 0, stream>>>(ws + OF_XC + I_SIZE, CATW,
                                                I_SIZE, ff_w, ff_b, ws + OF_L1);
  // kin: logits = [Xi | Xh] @ kin_w^T + kin_b -> L2
  k_gemm16<<<dim3(256), dim3(256), 0, stream>>>(ws + OF_XC, CATW,
                                                2*I_SIZE, kin_w, kin_b, ws + OF_L2);
  k_pred_kld<<<dim3(NB), dim3(1024), 0, stream>>>(ws);
  k_softmax_newinp<<<dim3(NB), dim3(1024), 0, stream>>>(ws);
  // kouth -> L1 (reuse), kouti -> L2 (reuse)
  k_gemm16<<<dim3(256), dim3(256), 0, stream>>>(ws + OF_NEWI, I_SIZE,
                                                I_SIZE, kouth_w, kouth_b, ws + OF_L1);
  k_gemm16<<<dim3(256), dim3(256), 0, stream>>>(ws + OF_NEWI, I_SIZE,
                                                I_SIZE, kouti_w, kouti_b, ws + OF_L2);
  k_proj_kld<<<dim3(2*NB), dim3(1024), 0, stream>>>(ws);
  k_gate<<<dim3(1), dim3(256), 0, stream>>>(gate_w, gate_b, ws, out);
  k_combine<<<dim3(H_SIZE), dim3(256), 0, stream>>>(stackM, input, ws, out);
}